// TSGBlock_38946763440921
// MI455X (gfx1250) — compile-verified
//
#include <hip/hip_runtime.h>
#include <hip/hip_bf16.h>

typedef float v2f __attribute__((ext_vector_type(2)));
typedef float v8f __attribute__((ext_vector_type(8)));

#define B_ 4
#define L_ 1024
#define D_ 512
#define N_ 16
#define CHUNK 64
#define NCHUNK 16   // L_/CHUNK
#define LOG2E 1.4426950408889634f

// ---------------------------------------------------------------------------
// K1: dbl[branch][b*L + l][48] = x[b,l,:] @ xproj_w^T   (M=4096, N=48, K=512)
// fp32 WMMA 16x16x4. One wave per 16x16 output tile.
// ---------------------------------------------------------------------------
__global__ void k_xproj(const float* __restrict__ x_rgb, const float* __restrict__ x_e,
                        const float* __restrict__ w1, const float* __restrict__ w2,
                        float* __restrict__ dbl) {
    int wave = (blockIdx.x * blockDim.x + threadIdx.x) >> 5;
    int ln = threadIdx.x & 31;
    // tiles: branch(2) x mtile(256) x ntile(3)
    int ntile = wave % 3;
    int rem   = wave / 3;
    int mtile = rem & 255;
    int branch = rem >> 8;
    const float* x = branch ? x_e : x_rgb;
    const float* w = branch ? w2 : w1;
    int m  = mtile * 16 + (ln & 15);
    int n  = ntile * 16 + (ln & 15);
    int kb = (ln >> 4) * 2;      // K sub-offset per half-wave
    const float* xrow = x + (long)m * D_;
    const float* wrow = w + (long)n * D_;
    v8f c = {};
    for (int k = 0; k < D_; k += 4) {
        v2f a, b;
        a.x = xrow[k + kb];  a.y = xrow[k + kb + 1];
        b.x = wrow[k + kb];  b.y = wrow[k + kb + 1];
        c = __builtin_amdgcn_wmma_f32_16x16x4_f32(false, a, false, b, (short)0, c, false, false);
    }
    float* out = dbl + (long)branch * 4096 * 48;
    int rowbase = mtile * 16 + ((ln >> 4) << 3);   // lanes>=16 hold M = i+8
    int col = ntile * 16 + (ln & 15);
#pragma unroll
    for (int i = 0; i < 8; ++i)
        out[(long)(rowbase + i) * 48 + col] = c[i];
}

// ---------------------------------------------------------------------------
// K2: delta_sp[branch][b*L+l][d] = softplus( dt[b,l,:] @ dtw[d,:]^T + dt_bias[d] )
// M=4096, N=512, K=16. fp32 WMMA, 4 K-steps, fused softplus epilogue.
// ---------------------------------------------------------------------------
__global__ void k_dtproj(const float* __restrict__ dbl,
                         const float* __restrict__ w1, const float* __restrict__ bias1,
                         const float* __restrict__ w2, const float* __restrict__ bias2,
                         float* __restrict__ delta_sp) {
    int wave = (blockIdx.x * blockDim.x + threadIdx.x) >> 5;
    int ln = threadIdx.x & 31;
    int ntile = wave & 31;
    int rem   = wave >> 5;
    int mtile = rem & 255;
    int branch = rem >> 8;
    const float* w    = branch ? w2 : w1;
    const float* bias = branch ? bias2 : bias1;
    const float* dt   = dbl + (long)branch * 4096 * 48;  // dt = first 16 cols of dbl
    int m  = mtile * 16 + (ln & 15);
    int n  = ntile * 16 + (ln & 15);
    int kb = (ln >> 4) * 2;
    v8f c = {};
#pragma unroll
    for (int k = 0; k < 16; k += 4) {
        v2f a, b;
        a.x = dt[(long)m * 48 + k + kb];  a.y = dt[(long)m * 48 + k + kb + 1];
        b.x = w[(long)n * 16 + k + kb];   b.y = w[(long)n * 16 + k + kb + 1];
        c = __builtin_amdgcn_wmma_f32_16x16x4_f32(false, a, false, b, (short)0, c, false, false);
    }
    float* out = delta_sp + (long)branch * 4096 * D_;
    int rowbase = mtile * 16 + ((ln >> 4) << 3);
    int col = ntile * 16 + (ln & 15);
    float bcol = bias[col];
#pragma unroll
    for (int i = 0; i < 8; ++i) {
        float v  = c[i] + bcol;
        float sp = (v > 20.f) ? v : log1pf(__expf(v));   // softplus, overflow-safe
        out[(long)(rowbase + i) * D_ + col] = sp;
    }
}

// ---------------------------------------------------------------------------
// K3: chunked scan pass 1. One thread per d channel, 16 states in registers.
// prod(dA) computed as exp2(A2[n] * sum(s)) -- no per-step product needed.
// grid: branch(2) x b(4) x dchunk(4) x lchunk(16) = 512 blocks of 128.
// ---------------------------------------------------------------------------
__global__ void k_scan1(const float* __restrict__ x_rgb, const float* __restrict__ x_e,
                        const float* __restrict__ dbl, const float* __restrict__ delta_sp,
                        const float* __restrict__ Alog1, const float* __restrict__ Alog2,
                        float* __restrict__ hpart, float* __restrict__ aprod) {
    int bid = blockIdx.x;
    int lc  = bid & 15; bid >>= 4;
    int dch = bid & 3;  bid >>= 2;
    int b   = bid & 3;
    int branch = bid >> 2;
    int tid = threadIdx.x;
    int l0 = lc * CHUNK;

    __shared__ float Bsh[CHUNK * 16];
    {
        const float* src = dbl + (long)branch * 4096 * 48 + (long)(b * L_ + l0) * 48 + 16;
        for (int i = tid; i < CHUNK * 16; i += blockDim.x)
            Bsh[i] = src[(long)(i >> 4) * 48 + (i & 15)];
    }
    __syncthreads();

    int d = dch * 128 + tid;
    const float* Alog = branch ? Alog2 : Alog1;
    const float* x    = branch ? x_e : x_rgb;
    const float* ds   = delta_sp + (long)branch * 4096 * D_;

    float A2[N_], h[N_];
#pragma unroll
    for (int n = 0; n < N_; ++n) {
        A2[n] = -__expf(Alog[(long)d * N_ + n]) * LOG2E;   // A * log2(e)
        h[n] = 0.f;
    }
    float ssum = 0.f;
    for (int ll = 0; ll < CHUNK; ++ll) {
        long idx = (long)(b * L_ + l0 + ll) * D_ + d;
        float s = ds[idx];
        float u = x[idx];
        float su = s * u;
        ssum += s;
#pragma unroll
        for (int n = 0; n < N_; ++n) {
            float dA = __builtin_amdgcn_exp2f(s * A2[n]);
            h[n] = fmaf(dA, h[n], su * Bsh[ll * 16 + n]);
        }
    }
    long off = ((long)(branch * 4 + b) * NCHUNK + lc) * (D_ * N_) + (long)d * N_;
#pragma unroll
    for (int n = 0; n < N_; ++n) {
        hpart[off + n] = h[n];
        aprod[off + n] = __builtin_amdgcn_exp2f(ssum * A2[n]);   // prod of dA over chunk
    }
}

// ---------------------------------------------------------------------------
// K4: carry propagation across chunks. One thread per (branch,b,d,n) = 65536.
// ---------------------------------------------------------------------------
__global__ void k_carry(const float* __restrict__ hpart, const float* __restrict__ aprod,
                        float* __restrict__ hin) {
    int idx = blockIdx.x * blockDim.x + threadIdx.x;   // 65536
    int dn = idx & (D_ * N_ - 1);
    int bb = idx >> 13;                                // branch*4 + b
    float h = 0.f;
    for (int c = 0; c < NCHUNK; ++c) {
        long pos = ((long)bb * NCHUNK + c) * (D_ * N_) + dn;
        hin[pos] = h;
        h = fmaf(aprod[pos], h, hpart[pos]);
    }
}

// ---------------------------------------------------------------------------
// K5: scan pass 2. Replay with carry-in, contract with the OTHER branch's C,
// add D*u, write y[branch][b,l,d] to d_out (coalesced over d).
// ---------------------------------------------------------------------------
__global__ void k_scan2(const float* __restrict__ x_rgb, const float* __restrict__ x_e,
                        const float* __restrict__ dbl, const float* __restrict__ delta_sp,
                        const float* __restrict__ Alog1, const float* __restrict__ Alog2,
                        const float* __restrict__ D1, const float* __restrict__ D2,
                        const float* __restrict__ hin, float* __restrict__ out) {
    int bid = blockIdx.x;
    int lc  = bid & 15; bid >>= 4;
    int dch = bid & 3;  bid >>= 2;
    int b   = bid & 3;
    int branch = bid >> 2;
    int tid = threadIdx.x;
    int l0 = lc * CHUNK;

    __shared__ float Bsh[CHUNK * 16];
    __shared__ float Csh[CHUNK * 16];
    {
        const float* srcB = dbl + (long)branch * 4096 * 48 + (long)(b * L_ + l0) * 48 + 16;
        const float* srcC = dbl + (long)(1 - branch) * 4096 * 48 + (long)(b * L_ + l0) * 48 + 32;
        for (int i = tid; i < CHUNK * 16; i += blockDim.x) {
            int r = i >> 4, cc = i & 15;
            Bsh[i] = srcB[(long)r * 48 + cc];
            Csh[i] = srcC[(long)r * 48 + cc];
        }
    }
    __syncthreads();

    int d = dch * 128 + tid;
    const float* Alog = branch ? Alog2 : Alog1;
    const float* Dp   = branch ? D2 : D1;
    const float* x    = branch ? x_e : x_rgb;
    const float* ds   = delta_sp + (long)branch * 4096 * D_;

    float A2[N_], h[N_];
    long off = ((long)(branch * 4 + b) * NCHUNK + lc) * (D_ * N_) + (long)d * N_;
#pragma unroll
    for (int n = 0; n < N_; ++n) {
        A2[n] = -__expf(Alog[(long)d * N_ + n]) * LOG2E;
        h[n] = hin[off + n];
    }
    float Dd = Dp[d];
    float* yout = out + (long)branch * 4096 * D_;
    for (int ll = 0; ll < CHUNK; ++ll) {
        long idx = (long)(b * L_ + l0 + ll) * D_ + d;
        float s = ds[idx];
        float u = x[idx];
        float su = s * u;
        float y = 0.f;
#pragma unroll
        for (int n = 0; n < N_; ++n) {
            float dA = __builtin_amdgcn_exp2f(s * A2[n]);
            h[n] = fmaf(dA, h[n], su * Bsh[ll * 16 + n]);
            y = fmaf(h[n], Csh[ll * 16 + n], y);
        }
        yout[idx] = fmaf(Dd, u, y);
    }
}

// ---------------------------------------------------------------------------
// K6: in-place layernorm over d (512). One block per (branch, b*L+l) row.
// var = E[x^2] - E[x]^2 to match jnp.var.
// ---------------------------------------------------------------------------
__global__ void k_ln(float* __restrict__ out,
                     const float* __restrict__ g1, const float* __restrict__ bb1,
                     const float* __restrict__ g2, const float* __restrict__ bb2) {
    int row = blockIdx.x;           // 0..8191 ; branch = row>>12
    int branch = row >> 12;
    float* p = out + (long)row * D_;
    int tid = threadIdx.x;
    float v0 = p[tid], v1 = p[tid + 256];
    __shared__ float rs[256], rq[256];
    rs[tid] = v0 + v1;
    rq[tid] = v0 * v0 + v1 * v1;
    __syncthreads();
    for (int st = 128; st > 0; st >>= 1) {
        if (tid < st) { rs[tid] += rs[tid + st]; rq[tid] += rq[tid + st]; }
        __syncthreads();
    }
    float mean = rs[0] * (1.f / 512.f);
    float var  = rq[0] * (1.f / 512.f) - mean * mean;
    float inv  = rsqrtf(var + 1e-5f);
    const float* g = branch ? g2 : g1;
    const float* bb = branch ? bb2 : bb1;
    p[tid]       = (v0 - mean) * inv * g[tid]       + bb[tid];
    p[tid + 256] = (v1 - mean) * inv * g[tid + 256] + bb[tid + 256];
}

// ---------------------------------------------------------------------------
extern "C" void kernel_launch(void* const* d_in, const int* in_sizes, int n_in,
                              void* d_out, int out_size, void* d_ws, size_t ws_size,
                              hipStream_t stream) {
    const float* x_rgb = (const float*)d_in[0];
    const float* x_e   = (const float*)d_in[1];
    const float* xw1   = (const float*)d_in[2];
    const float* xw2   = (const float*)d_in[3];
    const float* dtw1  = (const float*)d_in[4];
    const float* dtb1  = (const float*)d_in[5];
    const float* dtw2  = (const float*)d_in[6];
    const float* dtb2  = (const float*)d_in[7];
    const float* Al1   = (const float*)d_in[8];
    const float* Al2   = (const float*)d_in[9];
    const float* Dv1   = (const float*)d_in[10];
    const float* Dv2   = (const float*)d_in[11];
    const float* g1    = (const float*)d_in[12];
    const float* lb1   = (const float*)d_in[13];
    const float* g2    = (const float*)d_in[14];
    const float* lb2   = (const float*)d_in[15];
    float* out = (float*)d_out;

    float* ws    = (float*)d_ws;
    float* dbl   = ws;                          // 2*4096*48      = 393216
    float* delta = dbl + 2 * 4096 * 48;         // 2*4096*512     = 4194304
    float* hpart = delta + 2 * 4096 * 512;      // 2*4*16*8192    = 1048576
    float* aprod = hpart + 2 * 4 * NCHUNK * 8192;
    float* hin   = aprod + 2 * 4 * NCHUNK * 8192;

    // K1: 2*256*3 = 1536 waves, 4 waves/block -> 384 blocks
    k_xproj<<<384, 128, 0, stream>>>(x_rgb, x_e, xw1, xw2, dbl);
    // K2: 2*256*32 = 16384 waves -> 4096 blocks
    k_dtproj<<<4096, 128, 0, stream>>>(dbl, dtw1, dtb1, dtw2, dtb2, delta);
    // K3: 512 blocks x 128 threads
    k_scan1<<<512, 128, 0, stream>>>(x_rgb, x_e, dbl, delta, Al1, Al2, hpart, aprod);
    // K4: 65536 threads
    k_carry<<<256, 256, 0, stream>>>(hpart, aprod, hin);
    // K5: 512 blocks x 128 threads
    k_scan2<<<512, 128, 0, stream>>>(x_rgb, x_e, dbl, delta, Al1, Al2, Dv1, Dv2, hin, out);
    // K6: 2*4096 rows
    k_ln<<<8192, 256, 0, stream>>>(out, g1, lb1, g2, lb2);
}